// CAREGNN_5342939316746
// MI455X (gfx1250) — compile-verified
//
#include <hip/hip_runtime.h>
#include <hip/hip_bf16.h>

// ---------------------------------------------------------------------------
// CARE-GNN forward for MI455X (gfx1250, wave32).
//   h   = relu(x @ enc_w + enc_b)                     -> WMMA fp32 16x16x4
//   2x  care layer: per-node gates + per-edge sigmoid-gated mean aggregation
//   out = h @ cls_w + cls_b
// ---------------------------------------------------------------------------

typedef float v2f __attribute__((ext_vector_type(2)));
typedef float v8f __attribute__((ext_vector_type(8)));

#define N_NODES 50000
#define IN_DIM  256
#define HID     128

// ---------------------------------------------------------------------------
// Kernel 1: encoder GEMM with V_WMMA_F32_16X16X4_F32.
// Block = 256 threads = 8 waves. blockIdx.x selects a 16-row tile of x;
// wave w owns output columns [16w, 16w+16) (HID = 128 = 8 * 16).
// A fragment (16x4 f32, 2 VGPR): lanes 0-15 rows M=0..15 K={k,k+1},
//                                lanes 16-31 rows M=0..15 K={k+2,k+3}.
// B fragment (4x16 f32, 2 VGPR): same K split, lanes index column N.
// C/D (16x16 f32, 8 VGPR): VGPR r -> row r + 8*(lane>=16), col = lane&15.
// ---------------------------------------------------------------------------
__global__ __launch_bounds__(256) void encoder_wmma(
    const float* __restrict__ x, const float* __restrict__ w,
    const float* __restrict__ bias, float* __restrict__ h) {
  const int lane  = threadIdx.x & 31;
  const int wave  = threadIdx.x >> 5;
  const int m0    = blockIdx.x * 16;          // 50000 / 16 = 3125 exact
  const int n0    = wave * 16;
  const int idx16 = lane & 15;                // row for A, col for B
  const int khalf = (lane >> 4) * 2;          // 0 or 2

  const float* xr = x + (size_t)(m0 + idx16) * IN_DIM;
  v8f acc = {};
  for (int k = 0; k < IN_DIM; k += 4) {
    v2f a, b;
    a.x = xr[k + khalf];
    a.y = xr[k + khalf + 1];
    b.x = w[(size_t)(k + khalf)     * HID + n0 + idx16];
    b.y = w[(size_t)(k + khalf + 1) * HID + n0 + idx16];
    // 8 args: (neg_a, A, neg_b, B, c_mod, C, reuse_a, reuse_b)
    acc = __builtin_amdgcn_wmma_f32_16x16x4_f32(
        false, a, false, b, (short)0, acc, false, false);
  }

  const int orow = (lane >> 4) * 8;
  const int col  = lane & 15;
  const float bn = bias[n0 + col];
#pragma unroll
  for (int r = 0; r < 8; ++r) {
    float v = acc[r] + bn;
    h[(size_t)(m0 + orow + r) * HID + n0 + col] = fmaxf(v, 0.0f);
  }
}

// ---------------------------------------------------------------------------
// Kernel 2: per-node attention gates. gi[n] = h[n] . att_w[0:128],
// gj[n] = h[n] . att_w[128:256]. One wave per node; lane l covers 4 channels.
// ---------------------------------------------------------------------------
__global__ __launch_bounds__(256) void gates_kernel(
    const float* __restrict__ h, const float* __restrict__ att_w,
    float* __restrict__ gi, float* __restrict__ gj, int n_nodes) {
  const int node = (int)((blockIdx.x * (unsigned)blockDim.x + threadIdx.x) >> 5);
  const int lane = threadIdx.x & 31;
  if (node >= n_nodes) return;

  const float4 hv = ((const float4*)(h + (size_t)node * HID))[lane];
  const float4 wi = ((const float4*)att_w)[lane];
  const float4 wj = ((const float4*)(att_w + HID))[lane];

  float si = hv.x * wi.x + hv.y * wi.y + hv.z * wi.z + hv.w * wi.w;
  float sj = hv.x * wj.x + hv.y * wj.y + hv.z * wj.z + hv.w * wj.w;
#pragma unroll
  for (int off = 16; off > 0; off >>= 1) {
    si += __shfl_down(si, off, 32);
    sj += __shfl_down(sj, off, 32);
  }
  if (lane == 0) { gi[node] = si; gj[node] = sj; }
}

// ---------------------------------------------------------------------------
// Kernel 3: edge scatter. One wave per edge. alpha is wave-uniform
// (scalarizable); each lane gates 4 channels of h[src] and float-atomic-adds
// them into sums[dst] (resolves in L2: working set << 192 MB).
// ---------------------------------------------------------------------------
__global__ __launch_bounds__(256) void scatter_kernel(
    const float* __restrict__ h, const int* __restrict__ src,
    const int* __restrict__ dst, const float* __restrict__ gi,
    const float* __restrict__ gj, const float* __restrict__ att_b,
    float* __restrict__ sums, float* __restrict__ cnt, int n_edges) {
  const int e    = (int)((blockIdx.x * (unsigned)blockDim.x + threadIdx.x) >> 5);
  const int lane = threadIdx.x & 31;
  if (e >= n_edges) return;

  const int s = src[e];
  const int d = dst[e];
  float a = gi[d] + gj[s] + att_b[0];
  a = 1.0f / (1.0f + __expf(-a));           // sigmoid

  const float4 hs = ((const float4*)(h + (size_t)s * HID))[lane];
  float* sp = sums + (size_t)d * HID + lane * 4;
  atomicAdd(sp + 0, a * hs.x);
  atomicAdd(sp + 1, a * hs.y);
  atomicAdd(sp + 2, a * hs.z);
  atomicAdd(sp + 3, a * hs.w);
  if (lane == 0) atomicAdd(cnt + d, 1.0f);
}

// ---------------------------------------------------------------------------
// Kernel 4: mean + ReLU, written back over h (layer-1 epilogue).
// ---------------------------------------------------------------------------
__global__ __launch_bounds__(256) void finalize_relu(
    const float* __restrict__ sums, const float* __restrict__ cnt,
    float* __restrict__ h, int total) {
  const int i = blockIdx.x * blockDim.x + threadIdx.x;
  if (i >= total) return;
  const float c = fmaxf(cnt[i >> 7], 1.0f);
  h[i] = fmaxf(sums[i] / c, 0.0f);
}

// ---------------------------------------------------------------------------
// Kernel 5: layer-2 epilogue fused with classifier. One wave per node:
// mean, then 128x2 dot, shfl-reduce, lane 0 writes both logits.
// ---------------------------------------------------------------------------
__global__ __launch_bounds__(256) void finalize_cls(
    const float* __restrict__ sums, const float* __restrict__ cnt,
    const float* __restrict__ cls_w, const float* __restrict__ cls_b,
    float* __restrict__ out, int n_nodes) {
  const int node = (int)((blockIdx.x * (unsigned)blockDim.x + threadIdx.x) >> 5);
  const int lane = threadIdx.x & 31;
  if (node >= n_nodes) return;

  const float inv = 1.0f / fmaxf(cnt[node], 1.0f);
  const float4 sv = ((const float4*)(sums + (size_t)node * HID))[lane];
  const int c0 = lane * 4;
  float h0 = sv.x * inv, h1 = sv.y * inv, h2 = sv.z * inv, h3 = sv.w * inv;

  float o0 = h0 * cls_w[(c0 + 0) * 2 + 0] + h1 * cls_w[(c0 + 1) * 2 + 0] +
             h2 * cls_w[(c0 + 2) * 2 + 0] + h3 * cls_w[(c0 + 3) * 2 + 0];
  float o1 = h0 * cls_w[(c0 + 0) * 2 + 1] + h1 * cls_w[(c0 + 1) * 2 + 1] +
             h2 * cls_w[(c0 + 2) * 2 + 1] + h3 * cls_w[(c0 + 3) * 2 + 1];
#pragma unroll
  for (int off = 16; off > 0; off >>= 1) {
    o0 += __shfl_down(o0, off, 32);
    o1 += __shfl_down(o1, off, 32);
  }
  if (lane == 0) {
    out[(size_t)node * 2 + 0] = o0 + cls_b[0];
    out[(size_t)node * 2 + 1] = o1 + cls_b[1];
  }
}

// ---------------------------------------------------------------------------
extern "C" void kernel_launch(void* const* d_in, const int* in_sizes, int n_in,
                              void* d_out, int out_size, void* d_ws,
                              size_t ws_size, hipStream_t stream) {
  const float* x      = (const float*)d_in[0];
  const int*   eidx   = (const int*)d_in[1];
  const float* enc_w  = (const float*)d_in[2];
  const float* enc_b  = (const float*)d_in[3];
  const float* att1_w = (const float*)d_in[4];
  const float* att1_b = (const float*)d_in[5];
  const float* att2_w = (const float*)d_in[6];
  const float* att2_b = (const float*)d_in[7];
  const float* cls_w  = (const float*)d_in[8];
  const float* cls_b  = (const float*)d_in[9];
  float*       out    = (float*)d_out;

  const int n_nodes = in_sizes[0] / IN_DIM;      // 50000
  const int n_edges = in_sizes[1] / 2;           // 1.6M
  const int* src = eidx;
  const int* dst = eidx + n_edges;

  // Workspace layout (256-aligned):
  //   h    : n_nodes*HID f32   (25.6 MB)
  //   sums : n_nodes*HID f32   (25.6 MB)
  //   gi, gj, cnt : n_nodes f32 each (256 KB slots)
  const size_t NODE_F = (size_t)n_nodes * HID * sizeof(float);
  const size_t SLOT   = 262144;                  // >= 50000*4, 256-aligned
  char* ws = (char*)d_ws;
  float* h    = (float*)(ws);
  float* sums = (float*)(ws + NODE_F);
  float* gi   = (float*)(ws + 2 * NODE_F);
  float* gj   = (float*)(ws + 2 * NODE_F + SLOT);
  float* cnt  = (float*)(ws + 2 * NODE_F + 2 * SLOT);

  const int waveBlocksNodes = (n_nodes * 32 + 255) / 256;   // wave-per-node grids
  const int waveBlocksEdges = (n_edges * 32 + 255) / 256;   // wave-per-edge grids
  const int elemBlocks      = (n_nodes * HID + 255) / 256;

  // ---- encoder: h = relu(x @ enc_w + enc_b) via WMMA fp32 ----
  encoder_wmma<<<n_nodes / 16, 256, 0, stream>>>(x, enc_w, enc_b, h);

  // ---- CARE layer 1 ----
  gates_kernel<<<waveBlocksNodes, 256, 0, stream>>>(h, att1_w, gi, gj, n_nodes);
  hipMemsetAsync(sums, 0, NODE_F, stream);
  hipMemsetAsync(cnt, 0, (size_t)n_nodes * sizeof(float), stream);
  scatter_kernel<<<waveBlocksEdges, 256, 0, stream>>>(h, src, dst, gi, gj,
                                                      att1_b, sums, cnt, n_edges);
  finalize_relu<<<elemBlocks, 256, 0, stream>>>(sums, cnt, h, n_nodes * HID);

  // ---- CARE layer 2 + classifier ----
  gates_kernel<<<waveBlocksNodes, 256, 0, stream>>>(h, att2_w, gi, gj, n_nodes);
  hipMemsetAsync(sums, 0, NODE_F, stream);
  hipMemsetAsync(cnt, 0, (size_t)n_nodes * sizeof(float), stream);
  scatter_kernel<<<waveBlocksEdges, 256, 0, stream>>>(h, src, dst, gi, gj,
                                                      att2_b, sums, cnt, n_edges);
  finalize_cls<<<waveBlocksNodes, 256, 0, stream>>>(sums, cnt, cls_w, cls_b,
                                                    out, n_nodes);
}